// MoEGate_35107062678428
// MI455X (gfx1250) — compile-verified
//
#include <hip/hip_runtime.h>

typedef __attribute__((ext_vector_type(2))) float v2f;
typedef __attribute__((ext_vector_type(8))) float v8f;
typedef __attribute__((ext_vector_type(4))) int   v4i;

#define HDIM 4096
#define NEXP 64
#define KT   64           // K tile
#define TPW  128          // tokens per workgroup
#define XSTRIDE 68        // padded LDS row stride (floats): 16B-aligned rows, conflict-free b64 frags
#define NITER (HDIM / KT) // 64
#define NASYNC 12         // async-to-LDS instructions issued per wave per tile (8 X + 4 W)

typedef __attribute__((address_space(1))) v4i* g4_t;   // global int4* (AS1)
typedef __attribute__((address_space(3))) v4i* l4_t;   // LDS int4* (AS3)

__global__ __launch_bounds__(256, 1)
void moe_gate_wmma(const float* __restrict__ X, const float* __restrict__ W,
                   float* __restrict__ outw, int* __restrict__ outi)
{
    // Double-buffered tiles: X 2x128x68, W 2x64x68  -> 104448 B LDS
    __shared__ float ldsX[2][TPW * XSTRIDE];
    __shared__ float ldsW[2][NEXP * XSTRIDE];

    const int tid   = threadIdx.x;
    const int lane  = tid & 31;
    const int wave  = tid >> 5;
    const int half  = lane >> 4;      // 0: lanes 0-15, 1: lanes 16-31
    const int l16   = lane & 15;
    const int wgTok = blockIdx.x * TPW;

    v8f acc0 = {}, acc1 = {}, acc2 = {}, acc3 = {};

    // Async DMA of one K-tile into LDS buffer `buf` (per-lane LDS dest, 16B aligned)
    auto issue_tile = [&](int buf, int k0) {
        #pragma unroll
        for (int j = 0; j < 8; ++j) {               // X: 128x64 floats = 2048 b128, 8/thread
            int f4  = tid + 256 * j;
            int row = f4 >> 4;
            int c4  = (f4 & 15) << 2;
            const float* g = X + (size_t)(wgTok + row) * HDIM + k0 + c4;
            float*       l = &ldsX[buf][row * XSTRIDE + c4];
            __builtin_amdgcn_global_load_async_to_lds_b128((g4_t)g, (l4_t)l, 0, 0);
        }
        #pragma unroll
        for (int j = 0; j < 4; ++j) {               // W: 64x64 floats = 1024 b128, 4/thread
            int f4  = tid + 256 * j;
            int row = f4 >> 4;
            int c4  = (f4 & 15) << 2;
            const float* g = W + (size_t)row * HDIM + k0 + c4;
            float*       l = &ldsW[buf][row * XSTRIDE + c4];
            __builtin_amdgcn_global_load_async_to_lds_b128((g4_t)g, (l4_t)l, 0, 0);
        }
    };

    issue_tile(0, 0);   // prologue: fill buffer 0

    for (int i = 0; i < NITER; ++i) {
        const int cur = i & 1;

        // Overlap: start DMA of next tile into the other buffer, then wait only
        // for the CURRENT tile (async ops retire in order: <=NASYNC outstanding
        // means tile i has landed, tile i+1 still streaming).
        if (i + 1 < NITER) {
            issue_tile(cur ^ 1, (i + 1) * KT);
            __builtin_amdgcn_s_wait_asynccnt(NASYNC);
        } else {
            __builtin_amdgcn_s_wait_asynccnt(0);
        }
        __syncthreads();   // publish tile across waves

        // Fragment pointers into the current buffer.
        // A (16x4 f32): lanes 0-15 K={k,k+1}, lanes 16-31 K={k+2,k+3}, M = l16 (own 16 rows)
        // B (4x16 f32): same K split, N = l16 (expert within 16-wide tile)
        const float* xf = &ldsX[cur][(wave * 16 + l16) * XSTRIDE + (half << 1)];
        const float* wf = &ldsW[cur][l16 * XSTRIDE + (half << 1)];

        #pragma unroll
        for (int kk = 0; kk < KT; kk += 4) {
            v2f a  = *(const v2f*)(xf + kk);
            v2f b0 = *(const v2f*)(wf + kk);
            v2f b1 = *(const v2f*)(wf + 16 * XSTRIDE + kk);
            v2f b2 = *(const v2f*)(wf + 32 * XSTRIDE + kk);
            v2f b3 = *(const v2f*)(wf + 48 * XSTRIDE + kk);
            acc0 = __builtin_amdgcn_wmma_f32_16x16x4_f32(false, a, false, b0, (short)0, acc0, false, false);
            acc1 = __builtin_amdgcn_wmma_f32_16x16x4_f32(false, a, false, b1, (short)0, acc1, false, false);
            acc2 = __builtin_amdgcn_wmma_f32_16x16x4_f32(false, a, false, b2, (short)0, acc2, false, false);
            acc3 = __builtin_amdgcn_wmma_f32_16x16x4_f32(false, a, false, b3, (short)0, acc3, false, false);
        }
        __syncthreads();   // all waves done reading before this buffer is re-filled (tile i+2)
    }

    // ---- spill logits to (now free) buffer-0 X region: wave-local 16 x 64 (stride 68) ----
    // C/D layout: VGPR r -> token row (r + 8*half), lane l16 -> expert column within tile
    float* lg = &ldsX[0][wave * 16 * XSTRIDE];
    #pragma unroll
    for (int r = 0; r < 8; ++r) {
        int tr = r + (half << 3);
        lg[tr * XSTRIDE +  0 + l16] = acc0[r];
        lg[tr * XSTRIDE + 16 + l16] = acc1[r];
        lg[tr * XSTRIDE + 32 + l16] = acc2[r];
        lg[tr * XSTRIDE + 48 + l16] = acc3[r];
    }
    __syncthreads();

    // ---- per-token softmax + top-8 + renorm (one token per lane, lanes 0..15) ----
    if (lane < 16) {
        float* row = &ldsX[0][(wave * 16 + lane) * XSTRIDE];
        float m = -__builtin_inff();
        #pragma unroll
        for (int i = 0; i < NEXP; ++i) m = fmaxf(m, row[i]);
        float S = 0.f;
        #pragma unroll
        for (int i = 0; i < NEXP; ++i) S += __expf(row[i] - m);

        float pv[8];
        int   pi[8];
        float denom = 0.f;
        #pragma unroll
        for (int j = 0; j < 8; ++j) {
            float best = -__builtin_inff();
            int bi = 0;
            for (int i = 0; i < NEXP; ++i) {
                float v = row[i];
                if (v > best) { best = v; bi = i; }
            }
            row[bi] = -__builtin_inff();
            float p = __expf(best - m) / S;   // softmax prob of selected expert
            pv[j] = p; pi[j] = bi; denom += p;
        }
        denom += 1e-20f;

        int tok = wgTok + wave * 16 + lane;
        #pragma unroll
        for (int j = 0; j < 8; ++j) {
            outw[(size_t)tok * 8 + j] = pv[j] / denom;
            outi[(size_t)tok * 8 + j] = pi[j];
        }
    }
}

extern "C" void kernel_launch(void* const* d_in, const int* in_sizes, int n_in,
                              void* d_out, int out_size, void* d_ws, size_t ws_size,
                              hipStream_t stream) {
    const float* X = (const float*)d_in[0];   // [4,4096,4096] f32 -> [16384,4096]
    const float* W = (const float*)d_in[1];   // [64,4096] f32
    const int T = in_sizes[0] / HDIM;         // 16384 tokens

    float* outw = (float*)d_out;                         // [T,8] f32 weights
    int*   outi = (int*)((float*)d_out + (size_t)T * 8); // [T,8] i32 indices

    dim3 grid(T / TPW), block(256);
    hipLaunchKernelGGL(moe_gate_wmma, grid, block, 0, stream, X, W, outw, outi);
}